// GINELayer_5317169512875
// MI455X (gfx1250) — compile-verified
//
#include <hip/hip_runtime.h>
#include <math.h>

typedef __attribute__((ext_vector_type(2))) float v2f;
typedef __attribute__((ext_vector_type(4))) float v4f;
typedef __attribute__((ext_vector_type(8))) float v8f;

#define HIDDEN 128
#define HALF_H 64

// ---------------------------------------------------------------------------
// Kernel 1: x = node_embeddings  (scatter target init; EPS==0 so x = node+agg)
// ---------------------------------------------------------------------------
__global__ __launch_bounds__(256)
void gine_init_x(const float* __restrict__ node, float* __restrict__ x, int nvec4) {
    int i = blockIdx.x * blockDim.x + threadIdx.x;
    int stride = gridDim.x * blockDim.x;
    const v4f* src = (const v4f*)node;
    v4f* dst = (v4f*)x;
    for (int j = i; j < nvec4; j += stride) dst[j] = src[j];
}

// ---------------------------------------------------------------------------
// Kernel 2: per-wave contiguous edge range; run-length accumulate (start_nodes
// sorted) and flush with float atomics only at segment boundaries.
// Lane owns columns [4*lane, 4*lane+4).
// ---------------------------------------------------------------------------
__global__ __launch_bounds__(256)
void gine_scatter(const float* __restrict__ node, const float* __restrict__ edge,
                  const int* __restrict__ start, const int* __restrict__ endn,
                  float* __restrict__ x, int E, int per_wave) {
    int wid = (blockIdx.x * blockDim.x + threadIdx.x) >> 5;
    int lane = threadIdx.x & 31;
    long e0 = (long)wid * per_wave;
    if (e0 >= E) return;
    long e1 = e0 + per_wave;
    if (e1 > E) e1 = E;

    int col = lane * 4;
    v4f acc; acc.x = 0.f; acc.y = 0.f; acc.z = 0.f; acc.w = 0.f;
    int cur = start[e0];

    for (long e = e0; e < e1; ++e) {
        int s = start[e];           // uniform across wave
        int d = endn[e];            // uniform across wave
        v4f ne = *(const v4f*)(node + (long)d * HIDDEN + col);
        v4f ee = __builtin_nontemporal_load((const v4f*)(edge + e * HIDDEN + col));
        v4f m;
        m.x = fmaxf(ne.x + ee.x, 0.f);
        m.y = fmaxf(ne.y + ee.y, 0.f);
        m.z = fmaxf(ne.z + ee.z, 0.f);
        m.w = fmaxf(ne.w + ee.w, 0.f);
        if (s != cur) {             // uniform branch: flush previous segment
            float* p = x + (long)cur * HIDDEN + col;
            atomicAdd(p + 0, acc.x);
            atomicAdd(p + 1, acc.y);
            atomicAdd(p + 2, acc.z);
            atomicAdd(p + 3, acc.w);
            acc = m;
            cur = s;
        } else {
            acc.x += m.x; acc.y += m.y; acc.z += m.z; acc.w += m.w;
        }
    }
    float* p = x + (long)cur * HIDDEN + col;
    atomicAdd(p + 0, acc.x);
    atomicAdd(p + 1, acc.y);
    atomicAdd(p + 2, acc.z);
    atomicAdd(p + 3, acc.w);
}

// ---------------------------------------------------------------------------
// Kernel 3: MLP  out = gelu(x@W1 + b1) @ W2 + b2 using V_WMMA_F32_16X16X4_F32.
// One wave per 16-row tile. ISA layouts:
//   A: lane l holds A[l%16][kb], A[l%16][kb+1], kb = kstep*4 + (l/16)*2
//   B: lane l holds B[kb][l%16], B[kb+1][l%16]
//   C/D: lane l, vgpr v holds D[v + 8*(l/16)][l%16]
// h is staged through per-wave LDS to convert D-layout -> A-layout.
// ---------------------------------------------------------------------------
__global__ __launch_bounds__(256)
void gine_mlp(const float* __restrict__ x,
              const float* __restrict__ W1, const float* __restrict__ b1,
              const float* __restrict__ W2, const float* __restrict__ b2,
              float* __restrict__ out, int n_nodes, int n_tiles) {
    __shared__ float hbuf[8][16 * HALF_H];   // 4 KB per wave, 32 KB total

    int wz   = threadIdx.x >> 5;
    int lane = threadIdx.x & 31;
    int t = blockIdx.x * 8 + wz;             // tile index (uniform per wave)
    if (t >= n_tiles) return;                // uniform exit: EXEC stays all-1s

    int m    = lane & 15;                    // A row / B,D column within tile
    int half = lane >> 4;                    // 0 or 1
    int row  = t * 16 + m;
    int arow = row < n_nodes ? row : (n_nodes - 1);
    const float* xrow = x + (long)arow * HIDDEN;

    // ---- Layer 1: [16,128] x [128,64] -> acc1[4] n-tiles of 16 ----
    v8f acc1[4] = {};
#pragma unroll 4
    for (int k = 0; k < 32; ++k) {
        int kb = k * 4 + half * 2;
        v2f a = *(const v2f*)(xrow + kb);
#pragma unroll
        for (int nt = 0; nt < 4; ++nt) {
            int c = nt * 16 + m;
            v2f b;
            b.x = W1[kb * HALF_H + c];
            b.y = W1[(kb + 1) * HALF_H + c];
            acc1[nt] = __builtin_amdgcn_wmma_f32_16x16x4_f32(
                false, a, false, b, (short)0, acc1[nt], false, false);
        }
    }

    // ---- bias + exact GELU -> LDS (D-layout write) ----
#pragma unroll
    for (int nt = 0; nt < 4; ++nt) {
        int c = nt * 16 + m;
        float bb = b1[c];
#pragma unroll
        for (int v = 0; v < 8; ++v) {
            int M = v + half * 8;
            float val = acc1[nt][v] + bb;
            float g = 0.5f * val * (1.0f + erff(val * 0.70710678118654752f));
            hbuf[wz][M * HALF_H + c] = g;
        }
    }
    // same-wave LDS RAW: DS pipe is in-order per wave; fence the compiler and
    // drain DScnt before re-reading as the A operand.
    asm volatile("s_wait_dscnt 0" ::: "memory");

    // ---- Layer 2: [16,64] x [64,128] -> acc2[8] n-tiles of 16 ----
    v8f acc2[8] = {};
#pragma unroll 4
    for (int k = 0; k < 16; ++k) {
        int kb = k * 4 + half * 2;
        v2f a;
        a.x = hbuf[wz][m * HALF_H + kb];
        a.y = hbuf[wz][m * HALF_H + kb + 1];
#pragma unroll
        for (int nt = 0; nt < 8; ++nt) {
            int c = nt * 16 + m;
            v2f b;
            b.x = W2[kb * HIDDEN + c];
            b.y = W2[(kb + 1) * HIDDEN + c];
            acc2[nt] = __builtin_amdgcn_wmma_f32_16x16x4_f32(
                false, a, false, b, (short)0, acc2[nt], false, false);
        }
    }

    // ---- bias + store (D-layout) ----
#pragma unroll
    for (int nt = 0; nt < 8; ++nt) {
        int c = nt * 16 + m;
        float bb = b2[c];
#pragma unroll
        for (int v = 0; v < 8; ++v) {
            int R = t * 16 + v + half * 8;
            if (R < n_nodes) out[(long)R * HIDDEN + c] = acc2[nt][v] + bb;
        }
    }
}

// ---------------------------------------------------------------------------
extern "C" void kernel_launch(void* const* d_in, const int* in_sizes, int n_in,
                              void* d_out, int out_size, void* d_ws, size_t ws_size,
                              hipStream_t stream) {
    const float* node  = (const float*)d_in[0];
    const float* edge  = (const float*)d_in[1];
    const int*   start = (const int*)d_in[2];
    const int*   endn  = (const int*)d_in[3];
    const float* W1    = (const float*)d_in[4];
    const float* b1    = (const float*)d_in[5];
    const float* W2    = (const float*)d_in[6];
    const float* b2    = (const float*)d_in[7];
    float* out = (float*)d_out;
    float* x   = (float*)d_ws;                 // [n_nodes, 128] accumulator

    int n_nodes = in_sizes[0] / HIDDEN;
    int E       = in_sizes[2];

    // 1) x = node
    int nvec4 = n_nodes * (HIDDEN / 4);
    gine_init_x<<<512, 256, 0, stream>>>(node, x, nvec4);

    // 2) scatter-add relu(node[end] + edge) into x[start]
    const int blocksB = 1024, threadsB = 256;
    int waves = blocksB * threadsB / 32;
    int per_wave = (E + waves - 1) / waves;
    gine_scatter<<<blocksB, threadsB, 0, stream>>>(node, edge, start, endn, x, E, per_wave);

    // 3) MLP via f32 WMMA
    int n_tiles = (n_nodes + 15) / 16;
    int blocksC = (n_tiles + 7) / 8;
    gine_mlp<<<blocksC, 256, 0, stream>>>(x, W1, b1, W2, b2, out, n_nodes, n_tiles);
}